// SurfEval_48713519071346
// MI455X (gfx1250) — compile-verified
//
#include <hip/hip_runtime.h>

// NURBS surface eval on gfx1250. Two-stage banded GEMM on the f32 WMMA pipe:
//   Stage A: Mid[b,d] = Bu (512x64, band-8) * C_d (64x64)          (8K WMMA)
//   Stage B: Tile     = Mid_tile (16x8 band) * Bv_tile^T (8x16)    (65K WMMA)
// then out = xyz / w.  All matrix math is V_WMMA_F32_16X16X4_F32 (full fp32).
// The cubic basis has support width 4 and the span index advances <= 2 across
// any 16-point grid tile, so an 8-wide gathered K window covers all nonzeros.

typedef float v2f __attribute__((ext_vector_type(2)));
typedef float v8f __attribute__((ext_vector_type(8)));

#define GRID_N 512
#define MCTRL  64
#define NCTRL  64
#define BATCH  8

// ---------------------------------------------------------------------------
// Densify the sparse cubic basis: Bu[g, uspan[g]-3+r] = Nu[g,r], rest zero.
// ---------------------------------------------------------------------------
__global__ void build_basis_kernel(const float* __restrict__ Nu,
                                   const int*   __restrict__ uspan,
                                   const float* __restrict__ Nv,
                                   const int*   __restrict__ vspan,
                                   float* __restrict__ Bu,
                                   float* __restrict__ Bv) {
  int t = blockIdx.x * blockDim.x + threadIdx.x;
  if (t < GRID_N) {
    float* row = Bu + (size_t)t * MCTRL;
    #pragma unroll
    for (int j = 0; j < MCTRL; ++j) row[j] = 0.0f;
    int s = uspan[t];
    #pragma unroll
    for (int r = 0; r < 4; ++r) row[s - 3 + r] = Nu[t * 4 + r];
  } else if (t < 2 * GRID_N) {
    int g = t - GRID_N;
    float* row = Bv + (size_t)g * NCTRL;
    #pragma unroll
    for (int j = 0; j < NCTRL; ++j) row[j] = 0.0f;
    int s = vspan[g];
    #pragma unroll
    for (int r = 0; r < 4; ++r) row[s - 3 + r] = Nv[g * 4 + r];
  }
}

__device__ __forceinline__ int band_min(int span) {
  int c = span - 3;
  return (c > 56) ? 56 : (c < 0 ? 0 : c);   // 8-wide window stays in [0,64)
}

// ---------------------------------------------------------------------------
// Stage A: Mid[(b*4+d)][g][n] = sum_m Bu[g][m] * ctrl[b][m][n][d]
// Block = (g-tile, batch), 4 waves, wave d = component. Banded K=8 -> 8 WMMA.
// ---------------------------------------------------------------------------
__global__ __launch_bounds__(128)
void stageA_kernel(const float* __restrict__ ctrl,   // (8,64,64,4)
                   const float* __restrict__ Bu,     // (512,64)
                   const int*   __restrict__ uspan,  // (512,)
                   float* __restrict__ Mid)          // (8,4,512,64)
{
  const int gt = blockIdx.x;
  const int bb = blockIdx.y;
  const int g0 = gt * 16;

  const int d    = threadIdx.x >> 5;
  const int lane = threadIdx.x & 31;
  const int hh   = lane >> 4;
  const int l    = lane & 15;

  const int umin = band_min(uspan[g0]);

  v8f acc0 = {}, acc1 = {}, acc2 = {}, acc3 = {};
  const float* BuRow = Bu + (size_t)(g0 + l) * MCTRL + umin;

  #pragma unroll
  for (int kb = 0; kb < 2; ++kb) {
    const int k0 = 4 * kb + 2 * hh;
    v2f a;
    a.x = BuRow[k0];
    a.y = BuRow[k0 + 1];
    const float* c0 = ctrl + ((size_t)(bb * MCTRL + umin + k0) * NCTRL) * 4 + d;
    const float* c1 = c0 + (size_t)NCTRL * 4;
    {
      v2f b; b.x = c0[(0 * 16 + l) * 4]; b.y = c1[(0 * 16 + l) * 4];
      acc0 = __builtin_amdgcn_wmma_f32_16x16x4_f32(false, a, false, b, (short)0, acc0, false, false);
    }
    {
      v2f b; b.x = c0[(1 * 16 + l) * 4]; b.y = c1[(1 * 16 + l) * 4];
      acc1 = __builtin_amdgcn_wmma_f32_16x16x4_f32(false, a, false, b, (short)0, acc1, false, false);
    }
    {
      v2f b; b.x = c0[(2 * 16 + l) * 4]; b.y = c1[(2 * 16 + l) * 4];
      acc2 = __builtin_amdgcn_wmma_f32_16x16x4_f32(false, a, false, b, (short)0, acc2, false, false);
    }
    {
      v2f b; b.x = c0[(3 * 16 + l) * 4]; b.y = c1[(3 * 16 + l) * 4];
      acc3 = __builtin_amdgcn_wmma_f32_16x16x4_f32(false, a, false, b, (short)0, acc3, false, false);
    }
  }

  // D-layout element (m, n): m = j + 8*hh, n = nb*16 + l.
  float* Mrow = Mid + ((size_t)(bb * 4 + d) * GRID_N + g0) * NCTRL;
  #pragma unroll
  for (int j = 0; j < 8; ++j) {
    const int m = j + 8 * hh;
    Mrow[(size_t)m * NCTRL +  0 + l] = acc0[j];
    Mrow[(size_t)m * NCTRL + 16 + l] = acc1[j];
    Mrow[(size_t)m * NCTRL + 32 + l] = acc2[j];
    Mrow[(size_t)m * NCTRL + 48 + l] = acc3[j];
  }
}

// ---------------------------------------------------------------------------
// Stage B: Tile[m][n] = sum_k Mid[b,d][g0+m][cmin+k] * Bv[h0+n][cmin+k]
// Block = (h-tile, g-tile, batch), wave d = component. Banded K=8 -> 2 WMMA.
// ---------------------------------------------------------------------------
__global__ __launch_bounds__(128)
void stageB_kernel(const float* __restrict__ Mid,    // (8,4,512,64)
                   const float* __restrict__ Bv,     // (512,64)
                   const int*   __restrict__ vspan,  // (512,)
                   float* __restrict__ out)          // (8,512,512,3)
{
  __shared__ float Sws[4][16][17];

  const int ht = blockIdx.x;
  const int gt = blockIdx.y;
  const int bb = blockIdx.z;
  const int g0 = gt * 16, h0 = ht * 16;

  const int d    = threadIdx.x >> 5;
  const int lane = threadIdx.x & 31;
  const int hh   = lane >> 4;
  const int l    = lane & 15;

  const int cmin = band_min(vspan[h0]);

  const float* Arow = Mid + ((size_t)(bb * 4 + d) * GRID_N + g0 + l) * NCTRL + cmin;
  const float* Brow = Bv + (size_t)(h0 + l) * NCTRL + cmin;

  v8f accO = {};
  #pragma unroll
  for (int kb = 0; kb < 2; ++kb) {
    const int k0 = 4 * kb + 2 * hh;
    v2f a, b;
    a.x = Arow[k0];
    a.y = Arow[k0 + 1];
    b.x = Brow[k0];
    b.y = Brow[k0 + 1];
    accO = __builtin_amdgcn_wmma_f32_16x16x4_f32(false, a, false, b, (short)0, accO, false, false);
  }

  #pragma unroll
  for (int j = 0; j < 8; ++j) Sws[d][j + 8 * hh][l] = accO[j];
  __syncthreads();

  // Rational divide + dense store (only HBM stream of the whole pipeline).
  for (int p = threadIdx.x; p < 256; p += 128) {
    const int m = p >> 4, n = p & 15;
    const float inv = 1.0f / Sws[3][m][n];
    const size_t o = (((size_t)bb * GRID_N + (g0 + m)) * GRID_N + (h0 + n)) * 3;
    out[o + 0] = Sws[0][m][n] * inv;
    out[o + 1] = Sws[1][m][n] * inv;
    out[o + 2] = Sws[2][m][n] * inv;
  }
}

// ---------------------------------------------------------------------------
// Fallback fused kernel (round-1 path): used only if ws can't hold Mid.
// ---------------------------------------------------------------------------
__global__ __launch_bounds__(128)
void surfeval_fused_kernel(const float* __restrict__ ctrl,
                           const float* __restrict__ Bu,
                           const float* __restrict__ Bv,
                           float* __restrict__ out)
{
  __shared__ float A1s[4][16][68];
  __shared__ float Sws[4][16][17];

  const int ht = blockIdx.x, gt = blockIdx.y, bb = blockIdx.z;
  const int g0 = gt * 16, h0 = ht * 16;
  const int d    = threadIdx.x >> 5;
  const int lane = threadIdx.x & 31;
  const int hh   = lane >> 4;
  const int l    = lane & 15;

  v8f acc0 = {}, acc1 = {}, acc2 = {}, acc3 = {};
  const float* BuRow = Bu + (size_t)(g0 + l) * MCTRL;
  #pragma unroll
  for (int kb = 0; kb < 16; ++kb) {
    const int k0 = 4 * kb + 2 * hh;
    v2f a; a.x = BuRow[k0]; a.y = BuRow[k0 + 1];
    const float* c0 = ctrl + ((size_t)(bb * MCTRL + k0) * NCTRL) * 4 + d;
    const float* c1 = c0 + (size_t)NCTRL * 4;
    { v2f b; b.x = c0[(0*16+l)*4]; b.y = c1[(0*16+l)*4];
      acc0 = __builtin_amdgcn_wmma_f32_16x16x4_f32(false, a, false, b, (short)0, acc0, false, false); }
    { v2f b; b.x = c0[(1*16+l)*4]; b.y = c1[(1*16+l)*4];
      acc1 = __builtin_amdgcn_wmma_f32_16x16x4_f32(false, a, false, b, (short)0, acc1, false, false); }
    { v2f b; b.x = c0[(2*16+l)*4]; b.y = c1[(2*16+l)*4];
      acc2 = __builtin_amdgcn_wmma_f32_16x16x4_f32(false, a, false, b, (short)0, acc2, false, false); }
    { v2f b; b.x = c0[(3*16+l)*4]; b.y = c1[(3*16+l)*4];
      acc3 = __builtin_amdgcn_wmma_f32_16x16x4_f32(false, a, false, b, (short)0, acc3, false, false); }
  }
  #pragma unroll
  for (int j = 0; j < 8; ++j) {
    const int m = j + 8 * hh;
    A1s[d][m][ 0 + l] = acc0[j];
    A1s[d][m][16 + l] = acc1[j];
    A1s[d][m][32 + l] = acc2[j];
    A1s[d][m][48 + l] = acc3[j];
  }
  __syncthreads();

  v8f accO = {};
  const float* BvRow = Bv + (size_t)(h0 + l) * NCTRL;
  #pragma unroll
  for (int kb = 0; kb < 16; ++kb) {
    const int k0 = 4 * kb + 2 * hh;
    v2f a, b;
    a.x = A1s[d][l][k0]; a.y = A1s[d][l][k0 + 1];
    b.x = BvRow[k0];     b.y = BvRow[k0 + 1];
    accO = __builtin_amdgcn_wmma_f32_16x16x4_f32(false, a, false, b, (short)0, accO, false, false);
  }
  #pragma unroll
  for (int j = 0; j < 8; ++j) Sws[d][j + 8 * hh][l] = accO[j];
  __syncthreads();

  for (int p = threadIdx.x; p < 256; p += 128) {
    const int m = p >> 4, n = p & 15;
    const float inv = 1.0f / Sws[3][m][n];
    const size_t o = (((size_t)bb * GRID_N + (g0 + m)) * GRID_N + (h0 + n)) * 3;
    out[o + 0] = Sws[0][m][n] * inv;
    out[o + 1] = Sws[1][m][n] * inv;
    out[o + 2] = Sws[2][m][n] * inv;
  }
}

extern "C" void kernel_launch(void* const* d_in, const int* in_sizes, int n_in,
                              void* d_out, int out_size, void* d_ws, size_t ws_size,
                              hipStream_t stream) {
  const float* ctrl  = (const float*)d_in[0];  // (8,64,64,4) f32
  const int*   uspan = (const int*)  d_in[1];  // (512,) i32
  const int*   vspan = (const int*)  d_in[2];  // (512,) i32
  const float* Nu    = (const float*)d_in[3];  // (512,4) f32
  const float* Nv    = (const float*)d_in[4];  // (512,4) f32

  float* Bu = (float*)d_ws;                    // 512*64 f32 = 128 KB
  float* Bv = Bu + (size_t)GRID_N * MCTRL;     // next 128 KB
  float* Mid = Bv + (size_t)GRID_N * NCTRL;    // 8*4*512*64 f32 = 4 MB

  const size_t need = ((size_t)2 * GRID_N * MCTRL +
                       (size_t)BATCH * 4 * GRID_N * NCTRL) * sizeof(float);

  build_basis_kernel<<<4, 256, 0, stream>>>(Nu, uspan, Nv, vspan, Bu, Bv);

  if (ws_size >= need) {
    dim3 gridA(GRID_N / 16, BATCH);                 // (32, 8)
    stageA_kernel<<<gridA, 128, 0, stream>>>(ctrl, Bu, uspan, Mid);
    dim3 gridB(GRID_N / 16, GRID_N / 16, BATCH);    // (32, 32, 8)
    stageB_kernel<<<gridB, 128, 0, stream>>>(Mid, Bv, vspan, (float*)d_out);
  } else {
    dim3 grid(GRID_N / 16, GRID_N / 16, BATCH);
    surfeval_fused_kernel<<<grid, 128, 0, stream>>>(ctrl, Bu, Bv, (float*)d_out);
  }
}